// Model_39676907883066
// MI455X (gfx1250) — compile-verified
//
#include <hip/hip_runtime.h>

typedef float v2f __attribute__((ext_vector_type(2)));
typedef float v4f __attribute__((ext_vector_type(4)));
typedef float v8f __attribute__((ext_vector_type(8)));

#define S_DIM 256
#define L_DIM 2048
#define H_DIM 16
#define C_DIM 128   /* L/H */

// One workgroup = (s, slab of 128 l-rows). 8 waves, each owns one 16-row M-tile.
__global__ __launch_bounds__(256) void attn_rank1_kernel(
    const float* __restrict__ q,
    const float* __restrict__ k,
    const float* __restrict__ v,
    float* __restrict__ out)
{
    __shared__ float v_lds[L_DIM];       // value matrix for this s, viewed (128 c', 16 h)
    __shared__ float q_lds[128];         // q slab (128 rows)
    __shared__ float k_lds[128];         // k slab (128 rows)
    __shared__ float tile[8][16][36];    // per-wave store-staging (pad 36 to break bank conflicts)

    const int s    = blockIdx.x;
    const int slab = blockIdx.y;         // 16 slabs x 128 rows = 2048
    const int t    = threadIdx.x;
    const int wave = t >> 5;
    const int lane = t & 31;

    // ---- stage v: 2048 floats via 512 x b128 loads ----
    const v4f* vg = (const v4f*)(v + (size_t)s * L_DIM);
    v4f* vl = (v4f*)v_lds;
    vl[t * 2 + 0] = vg[t * 2 + 0];
    vl[t * 2 + 1] = vg[t * 2 + 1];
    if (t < 128) {
        q_lds[t] = q[(size_t)s * L_DIM + slab * 128 + t];
        k_lds[t] = k[(size_t)s * L_DIM + slab * 128 + t];
    }
    __syncthreads();

    // ---- per-wave softmax attention tile (16x16), straight into WMMA A layout ----
    // A for V_WMMA_F32_16X16X4_F32 is 16x4 f32: lanes 0-15 -> M, half-wave selects K pair.
    const int M = lane & 15;
    const int h = lane >> 4;             // 0: K={4kc,4kc+1}, 1: K={4kc+2,4kc+3}
    const float qv = q_lds[wave * 16 + M];

    float kk[16];
    #pragma unroll
    for (int j = 0; j < 16; ++j) kk[j] = k_lds[wave * 16 + j];

    float mx = qv * kk[0];
    #pragma unroll
    for (int j = 1; j < 16; ++j) mx = fmaxf(mx, qv * kk[j]);

    float row[16];
    float Z = 0.0f;
    #pragma unroll
    for (int j = 0; j < 16; ++j) { row[j] = __expf(qv * kk[j] - mx); Z += row[j]; }
    const float rZ = 1.0f / Z;
    #pragma unroll
    for (int j = 0; j < 16; ++j) row[j] *= rZ;

    v2f a[4];
    #pragma unroll
    for (int kc = 0; kc < 4; ++kc) {
        a[kc].x = h ? row[4 * kc + 2] : row[4 * kc + 0];
        a[kc].y = h ? row[4 * kc + 3] : row[4 * kc + 1];
    }

    // ---- sweep N-tiles in pairs (32 consecutive c' per pass -> 128B-contiguous stores) ----
    const int c_lane = lane & 15;
    #pragma unroll
    for (int np = 0; np < 4; ++np) {
        const int n0 = np * 2;
        const int n1 = np * 2 + 1;
        v8f acc0 = {};
        v8f acc1 = {};
        #pragma unroll
        for (int kc = 0; kc < 4; ++kc) {
            const int kidx = 4 * kc + 2 * h;
            // B (4x16) tile: B[h_k][N] = v[(16n + N)*16 + h_k]; lane supplies its half-wave K pair.
            const float* p0 = &v_lds[(n0 * 16 + c_lane) * 16 + kidx];
            const float* p1 = &v_lds[(n1 * 16 + c_lane) * 16 + kidx];
            v2f b0, b1;
            b0.x = p0[0]; b0.y = p0[1];
            b1.x = p1[0]; b1.y = p1[1];
            // D = A x B + C, fp32, K=4 chunk
            acc0 = __builtin_amdgcn_wmma_f32_16x16x4_f32(
                false, a[kc], false, b0, (short)0, acc0, false, false);
            acc1 = __builtin_amdgcn_wmma_f32_16x16x4_f32(
                false, a[kc], false, b1, (short)0, acc1, false, false);
        }

        // D layout: VGPR r, lane L -> (M = r + 8*(L>>4), N = L&15). Re-tile through LDS
        // so each lane emits b128 stores on the (dominant) 268 MB output stream.
        #pragma unroll
        for (int r = 0; r < 8; ++r) {
            tile[wave][r + 8 * h][c_lane]      = acc0[r];
            tile[wave][r + 8 * h][16 + c_lane] = acc1[r];
        }
        // (wave-private LDS region: same-wave DS ops are in-order; compiler inserts dscnt waits)

        const int r2  = lane >> 1;           // output row within tile
        const int seg = lane & 1;            // which 16-float half of the 32-float row-pair
        const int l_row = slab * 128 + wave * 16 + r2;
        float* obase = out + ((size_t)s * L_DIM + l_row) * C_DIM + n0 * 16 + seg * 16;
        const float* trow = &tile[wave][r2][seg * 16];
        #pragma unroll
        for (int i = 0; i < 4; ++i) {
            v4f val;
            val.x = trow[4 * i + 0];
            val.y = trow[4 * i + 1];
            val.z = trow[4 * i + 2];
            val.w = trow[4 * i + 3];
            *(v4f*)(obase + 4 * i) = val;    // global_store_b128, 128B contiguous per row
        }
    }
}

extern "C" void kernel_launch(void* const* d_in, const int* in_sizes, int n_in,
                              void* d_out, int out_size, void* d_ws, size_t ws_size,
                              hipStream_t stream) {
    (void)in_sizes; (void)n_in; (void)out_size; (void)d_ws; (void)ws_size;
    const float* q = (const float*)d_in[0];
    const float* k = (const float*)d_in[1];
    const float* v = (const float*)d_in[2];
    // d_in[3] = key_padding_mask: reference reshapes it but never applies it.
    float* out = (float*)d_out;

    dim3 grid(S_DIM, 16, 1);
    dim3 block(256, 1, 1);
    attn_rank1_kernel<<<grid, block, 0, stream>>>(q, k, v, out);
}